// minSAD_transforms_layer_76020921139520
// MI455X (gfx1250) — compile-verified
//
#include <hip/hip_runtime.h>
#include <hip/hip_bf16.h>

// ---------------- problem constants (match reference) ----------------
#define NCAND 75
#define BT_   8
#define C_    3
#define H_    192
#define W_    192
#define CROP  174
#define H0_   9
#define W0_   9
#define TILE_ROWS 8
#define NTILES    22           // ceil(174/8)
#define LDS_ROWS  20           // max band is ~15 rows; 20 is safe
#define NPIX_INV  (1.0f / 90828.0f)   // 1/(3*174*174)

typedef float v2f __attribute__((ext_vector_type(2)));
typedef float v8f __attribute__((ext_vector_type(8)));
typedef int   vi4 __attribute__((vector_size(4 * sizeof(int))));   // b128 payload type

// ---------------- CDNA5 feature guards ----------------
#if defined(__HIP_DEVICE_COMPILE__) && defined(__gfx1250__) && \
    __has_builtin(__builtin_amdgcn_global_load_async_to_lds_b128)
#define USE_ASYNC 1
#else
#define USE_ASYNC 0
#endif

#if defined(__HIP_DEVICE_COMPILE__) && defined(__gfx1250__) && \
    __has_builtin(__builtin_amdgcn_wmma_f32_16x16x4_f32)
#define USE_WMMA 1
#else
#define USE_WMMA 0
#endif

#define AS1 __attribute__((address_space(1)))
#define AS3 __attribute__((address_space(3)))

__device__ __forceinline__ void wait_async0() {
#if defined(__HIP_DEVICE_COMPILE__) && defined(__gfx1250__)
#if __has_builtin(__builtin_amdgcn_s_wait_asynccnt)
  __builtin_amdgcn_s_wait_asynccnt(0);
#else
  asm volatile("s_wait_asynccnt 0" ::: "memory");
#endif
#endif
}

// Map output-crop coords (x,y) -> source sample coords (xs,ys), exactly per reference.
__device__ __forceinline__ void map_xy(int x, int y, float a_sw, float a_sh,
                                       float cr, float sr, float& xs, float& ys) {
  float aw = (float)(x + W0_) + a_sw;     // asc == 1 exactly
  float ah = (float)(y + H0_) + a_sh;
  const float cx = 95.5f, cy = 95.5f;     // (W-1)/2 = (H-1)/2
  float dw = aw - cx, dh = ah - cy;
  float awr = cr * dw - sr * dh;
  float ahr = sr * dw + cr * dh;
  float gx = awr / 95.5f;
  float gy = ahr / 95.5f;
  xs = ((gx + 1.0f) * 192.0f - 1.0f) * 0.5f;
  ys = ((gy + 1.0f) * 192.0f - 1.0f) * 0.5f;
}

// ---------------- kernel 1: per-(cand, frame, channel) SAD partial ----------------
__global__ __launch_bounds__(256) void sad_partial_kernel(
    const float* __restrict__ img, const float* __restrict__ ref,
    float* __restrict__ partial) {
  const int cand = blockIdx.x;
  const int bt   = blockIdx.y;
  const int c    = blockIdx.z;
  const int tid  = threadIdx.x;

  // candidate -> (-shift_h, -shift_w, -rot_rad); scale == 1
  const int ish  = cand / 15;
  const int isw  = (cand / 3) % 5;
  const int irot = cand % 3;
  const float a_sh  = 2.0f - (float)ish;
  const float a_sw  = 2.0f - (float)isw;
  const float a_rot = (float)(1 - irot) * 0.017453292519943295f;
  const float cr = cosf(a_rot);
  const float sr = sinf(a_rot);

  const float* plane  = img + (size_t)(bt * C_ + c) * (H_ * W_);
  const float* rplane = ref + (size_t)(bt * C_ + c) * (H_ * W_);

  __shared__ float tile[LDS_ROWS * W_];
  __shared__ float wred[8];

  float acc = 0.0f;

  for (int t = 0; t < NTILES; ++t) {
    const int y0   = t * TILE_ROWS;
    const int rows = min(TILE_ROWS, CROP - y0);

    // Exact input-row band for this tile (affine map -> extremes at corners).
    float d0, d1, d2, d3, dx;
    map_xy(0,        y0,          a_sw, a_sh, cr, sr, dx, d0);
    map_xy(CROP - 1, y0,          a_sw, a_sh, cr, sr, dx, d1);
    map_xy(0,        y0 + rows-1, a_sw, a_sh, cr, sr, dx, d2);
    map_xy(CROP - 1, y0 + rows-1, a_sw, a_sh, cr, sr, dx, d3);
    float ymn = fminf(fminf(d0, d1), fminf(d2, d3));
    float ymx = fmaxf(fmaxf(d0, d1), fmaxf(d2, d3));
    int ylo = (int)floorf(ymn) - 1;
    int yhi = (int)floorf(ymx) + 2;
    ylo = max(ylo, 0);
    yhi = min(yhi, H_ - 1);
    yhi = min(yhi, ylo + (LDS_ROWS - 1));
    const int nrows = yhi - ylo + 1;

    // Stage band [ylo..yhi] x full width: contiguous nrows*192 floats.
    const float* src  = plane + (size_t)ylo * W_;
    const int    nvec = nrows * (W_ / 4);     // float4 chunks
#if USE_ASYNC
    for (int i = tid; i < nvec; i += 256) {
      __builtin_amdgcn_global_load_async_to_lds_b128(
          (AS1 vi4*)(src + (size_t)i * 4),
          (AS3 vi4*)(tile + (size_t)i * 4), 0, 0);
    }
    wait_async0();
#else
    for (int i = tid; i < nvec; i += 256) {
      float4 v = ((const float4*)src)[i];
      ((float4*)tile)[i] = v;
    }
#endif
    __syncthreads();

    // Prefetch next tile's band into cache (global_prefetch_b8, probe-confirmed).
    {
      size_t pidx = (size_t)(yhi + 1) * W_ + (size_t)tid * 16;
      if (pidx < (size_t)(H_ * W_)) __builtin_prefetch(plane + pidx, 0, 1);
    }

    // Bilinear SAD over rows x 174 from LDS.
    const int npix = rows * CROP;
    for (int i = tid; i < npix; i += 256) {
      const int yy = i / CROP;
      const int xx = i - yy * CROP;
      const int oy = y0 + yy;
      float xs, ys;
      map_xy(xx, oy, a_sw, a_sh, cr, sr, xs, ys);
      float x0f = floorf(xs), y0f = floorf(ys);
      float wx = xs - x0f, wy = ys - y0f;
      int xi = (int)x0f, yi = (int)y0f;

      auto corner = [&](int cxi, int cyi, float w) -> float {
        bool valid = (cxi >= 0) & (cxi < W_) & (cyi >= 0) & (cyi < H_);
        float wv = valid ? w : 0.0f;
        int xc = min(max(cxi, 0), W_ - 1);
        int yl = min(max(cyi - ylo, 0), nrows - 1);
        return tile[yl * W_ + xc] * wv;
      };

      float val = corner(xi,     yi,     (1.0f - wx) * (1.0f - wy))
                + corner(xi + 1, yi,     wx          * (1.0f - wy))
                + corner(xi,     yi + 1, (1.0f - wx) * wy)
                + corner(xi + 1, yi + 1, wx          * wy);

      float r = rplane[(size_t)(H0_ + oy) * W_ + (W0_ + xx)];
      acc += fabsf(val - r);
    }
    __syncthreads();   // protect tile before next stage
  }

  // ---- wave reduction: f32 WMMA (A={acc,0}, B=ones) sums lane m with lane m+16 ----
  float s;
#if USE_WMMA
  {
    v2f A; A[0] = acc;  A[1] = 0.0f;
    v2f B; B[0] = 1.0f; B[1] = 1.0f;
    v8f Cm = {};
    v8f D = __builtin_amdgcn_wmma_f32_16x16x4_f32(
        false, A, false, B, (short)0, Cm, false, false);
    s = D[0] + D[1] + D[2] + D[3] + D[4] + D[5] + D[6] + D[7];
    s += __shfl_xor(s, 16, 32);     // rows 0-7 half + rows 8-15 half -> wave total
  }
#else
  s = acc;
  for (int m = 16; m >= 1; m >>= 1) s += __shfl_xor(s, m, 32);
#endif

  const int wid = tid >> 5, lane = tid & 31;
  if (lane == 0) wred[wid] = s;
  __syncthreads();
  if (tid == 0) {
    float tot = 0.0f;
    for (int w = 0; w < 8; ++w) tot += wred[w];
    partial[(cand * BT_ + bt) * C_ + c] = tot;
  }
}

// ---------------- kernel 2: combine channels, write sad + argmin ----------------
__global__ __launch_bounds__(256) void sad_finalize_kernel(
    const float* __restrict__ partial, float* __restrict__ out) {
  const int tid = threadIdx.x;
  for (int i = tid; i < NCAND * BT_; i += 256) {
    float v = (partial[i * 3 + 0] + partial[i * 3 + 1] + partial[i * 3 + 2]) * NPIX_INV;
    out[i] = v;
  }
  if (tid < BT_) {
    const int bt = tid;
    float best = 3.4e38f;
    int bestn = 0;
    for (int n = 0; n < NCAND; ++n) {
      const int i = n * BT_ + bt;
      float v = (partial[i * 3 + 0] + partial[i * 3 + 1] + partial[i * 3 + 2]) * NPIX_INV;
      if (v < best) { best = v; bestn = n; }   // first-min, matches jnp.argmin
    }
    int* oi = reinterpret_cast<int*>(out + NCAND * BT_);
    oi[0 * BT_ + bt] = bestn / 15;
    oi[1 * BT_ + bt] = (bestn / 3) % 5;
    oi[2 * BT_ + bt] = bestn % 3;
    oi[3 * BT_ + bt] = 0;
  }
}

extern "C" void kernel_launch(void* const* d_in, const int* in_sizes, int n_in,
                              void* d_out, int out_size, void* d_ws, size_t ws_size,
                              hipStream_t stream) {
  const float* img = (const float*)d_in[0];   // matrix  (1,8,3,192,192) f32
  const float* ref = (const float*)d_in[1];   // reference_matrix, same shape
  float* out     = (float*)d_out;             // 600 f32 sad + 32 i32 min_idx
  float* partial = (float*)d_ws;              // 1800 floats scratch

  dim3 grid(NCAND, BT_, C_);
  sad_partial_kernel<<<grid, 256, 0, stream>>>(img, ref, partial);
  sad_finalize_kernel<<<1, 256, 0, stream>>>(partial, out);
}